// CRFLayer_13950053778208
// MI455X (gfx1250) — compile-verified
//
#include <hip/hip_runtime.h>
#include <stdint.h>

#define TT 512
#define CC 48
#define CHUNK 8   // timesteps of potentials prefetched per async chunk

// ---- CDNA5 async global->LDS path (guarded; fallback = sync copy) ----
#if defined(__has_builtin)
#  if __has_builtin(__builtin_amdgcn_global_load_async_to_lds_b128) && \
      __has_builtin(__builtin_amdgcn_s_wait_asynccnt)
#    define HAVE_ASYNC 1
#  endif
#endif
#ifndef HAVE_ASYNC
#  define HAVE_ASYNC 0
#endif

typedef int v4i __attribute__((ext_vector_type(4)));

__device__ __forceinline__ void copy16_g2l(float* lds_dst, const float* gsrc) {
#if HAVE_ASYNC
  __builtin_amdgcn_global_load_async_to_lds_b128(
      (__attribute__((address_space(1))) v4i*)gsrc,
      (__attribute__((address_space(3))) v4i*)lds_dst,
      /*offset=*/0, /*cpol=*/0);
#else
  float4 v = *(const float4*)gsrc;
  *(float4*)lds_dst = v;
#endif
}

#if HAVE_ASYNC
#  define WAIT_ASYNC(n) __builtin_amdgcn_s_wait_asynccnt(n)
#else
#  define WAIT_ASYNC(n) ((void)0)
#endif

__global__ __launch_bounds__(32) void crf_viterbi_kernel(
    const float* __restrict__ pot,    // [B,T,C]
    const int*   __restrict__ slen,   // [B,1]
    const float* __restrict__ trans,  // [C,C] row-major: trans[i*C+j]
    float* __restrict__ out)          // [B,T,C] one-hot
{
  __shared__ float         s_trans[CC * CC];          // 9216 B
  __shared__ float         s_score[CC];               //  192 B
  __shared__ float         s_pot[2 * CHUNK * CC];     // 3072 B
  __shared__ unsigned char s_bp[TT * CC];             // 24576 B
  __shared__ unsigned char s_tag[TT];                 //  512 B

  const int b    = blockIdx.x;
  const int lane = threadIdx.x;

  int L = slen[b];
  if (L < 1)  L = 1;
  if (L > TT) L = TT;

  const float* potb = pot + (size_t)b * TT * CC;

  // Stage transitions into LDS: 2304 floats = 576 x b128, 18 per lane.
  #pragma unroll
  for (int q = 0; q < 18; ++q) {
    int e = lane + q * 32;
    copy16_g2l(&s_trans[e * 4], trans + e * 4);
  }

  // Prefetch one CHUNK of potentials rows [1+cc*CHUNK, 1+cc*CHUNK+CHUNK)
  // into buffer (cc&1). 8*48 floats = 96 x b128, 3 per lane. Rows are
  // clamped to T-1 so the per-wave ASYNCcnt stays uniform (clamped slots
  // are never read: computed rows are always <= L-1 <= T-1).
  auto prefetch = [&](int cc) {
    int rb = 1 + cc * CHUNK;
    float* pb = &s_pot[(cc & 1) * CHUNK * CC];
    #pragma unroll
    for (int q = 0; q < 3; ++q) {
      int e   = lane + q * 32;        // 0..95; each row = 12 b128 (no straddle)
      int row = rb + e / 12;
      if (row > TT - 1) row = TT - 1;
      int col = (e % 12) * 4;
      copy16_g2l(pb + e * 4, potb + (size_t)row * CC + col);
    }
  };
  prefetch(0);
  prefetch(1);
  WAIT_ASYNC(3);   // async loads complete in order: transitions + chunk0 done

  // Each lane owns tag j0 = lane and (for lanes 0..15) j1 = 32+lane.
  const int  j0   = lane;
  const int  j1   = 32 + (lane & 15);
  const bool has1 = lane < 16;

  // Pull this lane's two transition COLUMNS into VGPRs (96 regs).
  float ta[CC], tb[CC];
  #pragma unroll
  for (int i = 0; i < CC; ++i) {
    ta[i] = s_trans[i * CC + j0];
    tb[i] = s_trans[i * CC + j1];
  }

  // t = 0 init: score = potentials[:,0]
  s_score[j0] = potb[j0];
  if (has1) s_score[j1] = potb[j1];

  // ---- Forward Viterbi, stopping at seq_len (masked tail is identity) ----
  if (L > 1) {
    for (int c = 0;; ++c) {
      int rb = 1 + c * CHUNK;
      int re = rb + CHUNK;
      if (re > L) re = L;
      const float* pb = &s_pot[(c & 1) * CHUNK * CC];

      for (int t = rb; t < re; ++t) {
        const float* pr = pb + (t - rb) * CC;
        float m0, m1;
        int   a0, a1;
        {
          float s = s_score[0];
          m0 = s + ta[0]; a0 = 0;
          m1 = s + tb[0]; a1 = 0;
        }
        #pragma unroll
        for (int i = 1; i < CC; ++i) {
          float s  = s_score[i];        // LDS broadcast (same addr all lanes)
          float v0 = s + ta[i];
          if (v0 > m0) { m0 = v0; a0 = i; }   // strict > keeps FIRST argmax
          float v1 = s + tb[i];
          if (v1 > m1) { m1 = v1; a1 = i; }
        }
        // Single wave: all 48 loads above precede these stores in order.
        s_bp[t * CC + j0] = (unsigned char)a0;
        s_score[j0]       = m0 + pr[j0];
        if (has1) {
          s_bp[t * CC + j1] = (unsigned char)a1;
          s_score[j1]       = m1 + pr[j1];
        }
      }

      if (re >= L) break;
      prefetch(c + 2);   // outstanding: chunk c+1 (3) + chunk c+2 (3)
      WAIT_ASYNC(3);     // chunk c+1 landed; invariant restored
    }
  }

  // ---- argmax of final score (computed redundantly by every lane) ----
  float bm = s_score[0];
  int   bt = 0;
  #pragma unroll
  for (int i = 1; i < CC; ++i) {
    float s = s_score[i];
    if (s > bm) { bm = s; bt = i; }
  }

  // tags[t] = last_tag for t in [L-1, T): fill in parallel
  for (int t = L - 1 + lane; t < TT; t += 32) s_tag[t] = (unsigned char)bt;

  // Serial backtrace through LDS backpointers on lane 0
  if (lane == 0) {
    int tg = bt;
    for (int t = L - 1; t >= 1; --t) {
      tg = s_bp[t * CC + tg];
      s_tag[t - 1] = (unsigned char)tg;
    }
  }

  // ---- one-hot output: T*12 float4 groups, all 32 lanes, coalesced b128 ----
  float* outb = out + (size_t)b * TT * CC;
  for (int g = lane; g < TT * 12; g += 32) {
    int t  = g / 12;
    int c4 = (g % 12) * 4;
    int tg = s_tag[t];
    float4 v;
    v.x = (c4 + 0 == tg) ? 1.0f : 0.0f;
    v.y = (c4 + 1 == tg) ? 1.0f : 0.0f;
    v.z = (c4 + 2 == tg) ? 1.0f : 0.0f;
    v.w = (c4 + 3 == tg) ? 1.0f : 0.0f;
    *(float4*)(outb + (size_t)t * CC + c4) = v;
  }
}

extern "C" void kernel_launch(void* const* d_in, const int* in_sizes, int n_in,
                              void* d_out, int out_size, void* d_ws, size_t ws_size,
                              hipStream_t stream) {
  (void)n_in; (void)out_size; (void)d_ws; (void)ws_size;
  const float* pot   = (const float*)d_in[0];
  const int*   slen  = (const int*)d_in[1];
  const float* trans = (const float*)d_in[2];
  float*       out   = (float*)d_out;

  const int B = in_sizes[0] / (TT * CC);   // 1024
  crf_viterbi_kernel<<<dim3(B), dim3(32), 0, stream>>>(pot, slen, trans, out);
}